// BGNN_MLP_83614423318859
// MI455X (gfx1250) — compile-verified
//
#include <hip/hip_runtime.h>

#define DD 128

typedef __attribute__((ext_vector_type(2))) float v2f;
typedef __attribute__((ext_vector_type(8))) float v8f;

// C[N,128] = A[N,128] @ W[128,128] + bias, using fp32 WMMA 16x16x4.
// Block = 256 threads = 8 waves. Each block: 16 output rows, wave w -> cols [16w,16w+16).
__global__ __launch_bounds__(256) void bgnn_gemm_bias(
    const float* __restrict__ A,
    const float* __restrict__ W,
    const float* __restrict__ bias,
    float* __restrict__ C,
    int N)
{
    __shared__ float sW[DD * DD];   // 64 KB, row-major W
    const int tid = threadIdx.x;
    #pragma unroll
    for (int i = 0; i < (DD * DD) / 256; ++i) {
        const int idx = i * 256 + tid;
        sW[idx] = W[idx];
    }
    __syncthreads();

    const int wave = tid >> 5;
    const int lane = tid & 31;
    const int l16  = lane & 15;
    const int lh   = lane >> 4;          // 0: lanes 0-15, 1: lanes 16-31
    const int row0 = blockIdx.x * 16;
    const int col  = wave * 16 + l16;

    int arow = row0 + l16;
    if (arow >= N) arow = N - 1;         // clamp: keeps EXEC all-ones for WMMA
    const float* Ap = A + (size_t)arow * DD + 2 * lh;

    v8f acc = {};
    #pragma unroll
    for (int kb = 0; kb < DD; kb += 4) {
        v2f a, b;
        // A fragment 16x4: lanes 0-15 hold K=kb..kb+1 of row l16, lanes 16-31 hold K=kb+2..kb+3
        a.x = Ap[kb + 0];
        a.y = Ap[kb + 1];
        // B fragment 4x16: lanes 0-15 hold rows kb,kb+1 at column 'col'; lanes 16-31 rows kb+2,kb+3
        const float* wp = &sW[(kb + 2 * lh) * DD + col];
        b.x = wp[0];
        b.y = wp[DD];
        acc = __builtin_amdgcn_wmma_f32_16x16x4_f32(false, a, false, b,
                                                    (short)0, acc, false, false);
    }

    const float bv = bias[col];
    #pragma unroll
    for (int i = 0; i < 8; ++i) {
        const int row = row0 + lh * 8 + i;   // C/D layout: VGPR i -> M=i (+8 for upper half)
        if (row < N)
            C[(size_t)row * DD + col] = acc[i] + bv;
    }
}

// Hardware f32 atomic add, device scope, no return (STOREcnt-tracked).
__device__ __forceinline__ void gatomic_add_f32(float* p, float v, int byte_off) {
    switch (byte_off) {
    case 0:  asm volatile("global_atomic_add_f32 %0, %1, off scope:SCOPE_DEV"
                          :: "v"(p), "v"(v) : "memory"); break;
    case 4:  asm volatile("global_atomic_add_f32 %0, %1, off offset:4 scope:SCOPE_DEV"
                          :: "v"(p), "v"(v) : "memory"); break;
    case 8:  asm volatile("global_atomic_add_f32 %0, %1, off offset:8 scope:SCOPE_DEV"
                          :: "v"(p), "v"(v) : "memory"); break;
    default: asm volatile("global_atomic_add_f32 %0, %1, off offset:12 scope:SCOPE_DEV"
                          :: "v"(p), "v"(v) : "memory"); break;
    }
}

// dst[edst[e]] += src[esrc[e]] over 128-float rows.
// One float4 per lane: 32 lanes cover one edge row; gather is global_load_b128,
// updates are 4 coalesced global_atomic_add_f32 per lane (contiguous 512B per wave).
__global__ __launch_bounds__(256) void bgnn_scatter_add(
    const float4* __restrict__ src,   // [Nsrc, 32] float4
    const int* __restrict__ esrc,
    const int* __restrict__ edst,
    float* __restrict__ dst,          // [Ndst, 128] float
    int E)
{
    const long long gid = (long long)blockIdx.x * 256 + threadIdx.x;
    if (gid >= (long long)E * 32) return;
    const int e  = (int)(gid >> 5);
    const int c4 = (int)(gid & 31);

    const float4 v = src[(size_t)esrc[e] * 32 + c4];
    float* p = dst + (size_t)edst[e] * DD + c4 * 4;
    gatomic_add_f32(p, v.x, 0);
    gatomic_add_f32(p, v.y, 4);
    gatomic_add_f32(p, v.z, 8);
    gatomic_add_f32(p, v.w, 12);
}

extern "C" void kernel_launch(void* const* d_in, const int* in_sizes, int n_in,
                              void* d_out, int out_size, void* d_ws, size_t ws_size,
                              hipStream_t stream) {
    (void)n_in; (void)out_size; (void)ws_size;

    const float* X_v    = (const float*)d_in[1];
    const int*   edge_u = (const int*)d_in[2];
    const int*   edge_v = (const int*)d_in[3];
    const float* W0 = (const float*)d_in[4];
    const float* b0 = (const float*)d_in[5];
    const float* W1 = (const float*)d_in[6];
    const float* b1 = (const float*)d_in[7];
    const float* W2 = (const float*)d_in[8];
    const float* b2 = (const float*)d_in[9];
    float* out = (float*)d_out;

    const int NU = in_sizes[0] / DD;   // 50000
    const int NV = in_sizes[1] / DD;   // 50000
    const int E  = in_sizes[2];        // 640000
    const int NMAX = NU > NV ? NU : NV;

    // three ping-pong feature buffers in workspace (3 * 25.6 MB)
    float* bufA = (float*)d_ws;
    float* bufB = bufA + (size_t)NMAX * DD;
    float* bufC = bufB + (size_t)NMAX * DD;

    const dim3 blk(256);
    const unsigned sgrid = (unsigned)(((long long)E * 32 + 255) / 256);

    // layer 0: Y_v = X_v @ W0 + b0 ; U1 = v2u(Y_v)
    bgnn_gemm_bias<<<(NV + 15) / 16, blk, 0, stream>>>(X_v, W0, b0, bufA, NV);
    hipMemsetAsync(bufB, 0, (size_t)NU * DD * sizeof(float), stream);
    bgnn_scatter_add<<<sgrid, blk, 0, stream>>>((const float4*)bufA, edge_v, edge_u, bufB, E);

    // layer 1: Y_u = U1 @ W1 + b1 ; V1 = u2v(Y_u)
    bgnn_gemm_bias<<<(NU + 15) / 16, blk, 0, stream>>>(bufB, W1, b1, bufC, NU);
    hipMemsetAsync(bufA, 0, (size_t)NV * DD * sizeof(float), stream);
    bgnn_scatter_add<<<sgrid, blk, 0, stream>>>((const float4*)bufC, edge_u, edge_v, bufA, E);

    // layer 2: Y_v2 = V1 @ W2 + b2 ; out = v2u(Y_v2)
    bgnn_gemm_bias<<<(NV + 15) / 16, blk, 0, stream>>>(bufA, W2, b2, bufB, NV);
    hipMemsetAsync(out, 0, (size_t)NU * DD * sizeof(float), stream);
    bgnn_scatter_add<<<sgrid, blk, 0, stream>>>((const float4*)bufB, edge_v, edge_u, out, E);
}